// EncoderLayer_69423851373091
// MI455X (gfx1250) — compile-verified
//
#include <hip/hip_runtime.h>
#include <math.h>

#define NN   50000
#define EE   800000
#define EMB  128
#define HID  512
#define HH   8
#define DDIM 16
#define BN_EPS 1e-5f

typedef __attribute__((ext_vector_type(2))) float v2f;
typedef __attribute__((ext_vector_type(8))) float v8f;

#define LDS_STRIDE 132   // 128 + 4 pad -> conflict-free b64 reads across m

#define ATOMIC_ADD_F32(p, v) \
  __hip_atomic_fetch_add((p), (v), __ATOMIC_RELAXED, __HIP_MEMORY_SCOPE_AGENT)

__device__ __forceinline__ v8f wmma_f32(v2f a, v2f b, v8f c) {
#if defined(__gfx1250__)
  // V_WMMA_F32_16X16X4_F32 : D = A(16x4,f32) * B(4x16,f32) + C(16x16,f32)
  return __builtin_amdgcn_wmma_f32_16x16x4_f32(
      /*neg_a=*/false, a, /*neg_b=*/false, b,
      /*c_mod=*/(short)0, c, /*reuse_a=*/false, /*reuse_b=*/false);
#else
  return c;
#endif
}

// ---------------------------------------------------------------------------
// Kernel 0: zero a float region (grid-stride)
// ---------------------------------------------------------------------------
__global__ void zero_kernel(float* __restrict__ p, long n) {
  long i = (long)blockIdx.x * blockDim.x + threadIdx.x;
  long stride = (long)gridDim.x * blockDim.x;
  for (; i < n; i += stride) p[i] = 0.0f;
}

// ---------------------------------------------------------------------------
// Kernel 1: q,k,v = x @ {wq,wk,wv} + bias    (N x 128 @ 128 x 128, x3)
// 256 threads = 8 waves; block handles a 16-row M-tile; wave w -> cols 16w..16w+15
// ---------------------------------------------------------------------------
__global__ __launch_bounds__(256) void qkv_kernel(
    const float* __restrict__ x,
    const float* __restrict__ wq, const float* __restrict__ bq,
    const float* __restrict__ wk, const float* __restrict__ bk,
    const float* __restrict__ wv, const float* __restrict__ bv,
    float* __restrict__ q, float* __restrict__ k, float* __restrict__ v) {
  __shared__ __align__(16) float xs[16 * LDS_STRIDE];
  const int m0 = blockIdx.x * 16;
  const int t  = threadIdx.x;
  for (int i = t; i < 16 * EMB; i += 256) {
    int r = i >> 7, c = i & 127;
    xs[r * LDS_STRIDE + c] = x[(long)(m0 + r) * EMB + c];
  }
  __syncthreads();
  const int wave = t >> 5, lane = t & 31;
  const int n = wave * 16 + (lane & 15);   // output column
  const int m = lane & 15;                 // A row within tile
  const int ksel = (lane >> 4) * 2;        // A/B k sub-pair
  v8f cq = {}; v8f ck = {}; v8f cv = {};
  for (int kb = 0; kb < EMB; kb += 4) {
    v2f a = *(const v2f*)(xs + m * LDS_STRIDE + kb + ksel);
    const int kr = kb + ksel;
    v2f b0, b1, b2;
    b0[0] = wq[kr * EMB + n]; b0[1] = wq[(kr + 1) * EMB + n];
    b1[0] = wk[kr * EMB + n]; b1[1] = wk[(kr + 1) * EMB + n];
    b2[0] = wv[kr * EMB + n]; b2[1] = wv[(kr + 1) * EMB + n];
    cq = wmma_f32(a, b0, cq);
    ck = wmma_f32(a, b1, ck);
    cv = wmma_f32(a, b2, cv);
  }
  const float biasq = bq[n], biask = bk[n], biasv = bv[n];
  const int rbase = m0 + (lane >> 4) * 8;
  for (int j = 0; j < 8; ++j) {
    long idx = (long)(rbase + j) * EMB + n;
    q[idx] = cq[j] + biasq;
    k[idx] = ck[j] + biask;
    v[idx] = cv[j] + biasv;
  }
}

// ---------------------------------------------------------------------------
// Kernel 2: edge scatter: ksum[A[e]] += k[B[e]] ; cnt[A[e]] += 1
// 32 lanes per edge, 4 floats per lane (float4 gather, 4 hw f32 atomics)
// ---------------------------------------------------------------------------
__global__ __launch_bounds__(256) void edge_kernel(
    const int* __restrict__ eidx, const float* __restrict__ k,
    float* __restrict__ ksum, float* __restrict__ cnt) {
  long t = (long)blockIdx.x * 256 + threadIdx.x;
  long e = t >> 5;
  if (e >= EE) return;
  int sub = (int)(t & 31);
  int a = eidx[e];
  int b = eidx[EE + e];
  const float4 kv = *(const float4*)(k + (long)b * EMB + sub * 4);
  float* dst = ksum + (long)a * EMB + sub * 4;
  ATOMIC_ADD_F32(dst + 0, kv.x);
  ATOMIC_ADD_F32(dst + 1, kv.y);
  ATOMIC_ADD_F32(dst + 2, kv.z);
  ATOMIC_ADD_F32(dst + 3, kv.w);
  if (sub == 0) ATOMIC_ADD_F32(cnt + a, 1.0f);
}

// ---------------------------------------------------------------------------
// Kernel 3: per-node attention: score = (q[n] @ ksum[n]^T)/max(cnt,1);
//           softmax rows; attn[n] = score @ v[n].  128 threads per node.
// ---------------------------------------------------------------------------
__global__ __launch_bounds__(128) void attn_kernel(
    const float* __restrict__ q, const float* __restrict__ ksum,
    const float* __restrict__ v, const float* __restrict__ cnt,
    float* __restrict__ attn) {
  __shared__ float sq[EMB], sk[EMB], sv[EMB], ss[64], rmax[HH], rsum[HH];
  __shared__ float sinv;
  const int n = blockIdx.x;
  const int t = threadIdx.x;
  sq[t] = q[(long)n * EMB + t];
  sk[t] = ksum[(long)n * EMB + t];
  sv[t] = v[(long)n * EMB + t];
  if (t == 0) sinv = 1.0f / fmaxf(cnt[n], 1.0f);
  __syncthreads();
  if (t < 64) {
    int h = t >> 3, g = t & 7;
    float acc = 0.0f;
    for (int d = 0; d < DDIM; ++d) acc += sq[h * DDIM + d] * sk[g * DDIM + d];
    ss[t] = acc * sinv;
  }
  __syncthreads();
  if (t < HH) {
    float mx = ss[t * 8];
    for (int g = 1; g < 8; ++g) mx = fmaxf(mx, ss[t * 8 + g]);
    rmax[t] = mx;
  }
  __syncthreads();
  if (t < 64) ss[t] = expf(ss[t] - rmax[t >> 3]);
  __syncthreads();
  if (t < HH) {
    float s = 0.0f;
    for (int g = 0; g < 8; ++g) s += ss[t * 8 + g];
    rsum[t] = s;
  }
  __syncthreads();
  const int h = t >> 4, d = t & 15;
  float acc = 0.0f;
  for (int g = 0; g < 8; ++g) acc += ss[h * 8 + g] * sv[g * DDIM + d];
  attn[(long)n * EMB + t] = acc / rsum[h];
}

// ---------------------------------------------------------------------------
// Kernel 4: Y = A @ W(128x128) + bias + resid, accumulate column sum/sumsq
// ---------------------------------------------------------------------------
__global__ __launch_bounds__(256) void proj_res_kernel(
    const float* __restrict__ A, const float* __restrict__ W,
    const float* __restrict__ bias, const float* __restrict__ resid,
    float* __restrict__ Y, float* __restrict__ csum, float* __restrict__ csq) {
  __shared__ __align__(16) float xs[16 * LDS_STRIDE];
  const int m0 = blockIdx.x * 16;
  const int t  = threadIdx.x;
  for (int i = t; i < 16 * EMB; i += 256) {
    int r = i >> 7, c = i & 127;
    xs[r * LDS_STRIDE + c] = A[(long)(m0 + r) * EMB + c];
  }
  __syncthreads();
  const int wave = t >> 5, lane = t & 31;
  const int n = wave * 16 + (lane & 15);
  const int m = lane & 15;
  const int ksel = (lane >> 4) * 2;
  v8f c = {};
  for (int kb = 0; kb < EMB; kb += 4) {
    v2f a = *(const v2f*)(xs + m * LDS_STRIDE + kb + ksel);
    const int kr = kb + ksel;
    v2f b;
    b[0] = W[kr * EMB + n]; b[1] = W[(kr + 1) * EMB + n];
    c = wmma_f32(a, b, c);
  }
  const float bb = bias[n];
  const int rbase = m0 + (lane >> 4) * 8;
  float lsum = 0.0f, lsq = 0.0f;
  for (int j = 0; j < 8; ++j) {
    long idx = (long)(rbase + j) * EMB + n;
    float y = c[j] + bb + resid[idx];
    Y[idx] = y;
    lsum += y; lsq += y * y;
  }
  ATOMIC_ADD_F32(csum + n, lsum);
  ATOMIC_ADD_F32(csq + n, lsq);
}

// ---------------------------------------------------------------------------
// Kernel 5: batchnorm finalize (128 threads): scale/shift per column
// ---------------------------------------------------------------------------
__global__ void bn_finalize_kernel(
    const float* __restrict__ csum, const float* __restrict__ csq,
    const float* __restrict__ g, const float* __restrict__ be,
    float* __restrict__ scale, float* __restrict__ shift, float invn) {
  int c = threadIdx.x;
  float mu  = csum[c] * invn;
  float var = csq[c] * invn - mu * mu;
  float sc  = g[c] * rsqrtf(var + BN_EPS);
  scale[c] = sc;
  shift[c] = be[c] - mu * sc;
}

// ---------------------------------------------------------------------------
// Kernel 6: elementwise y*scale[col]+shift[col]
// ---------------------------------------------------------------------------
__global__ __launch_bounds__(256) void bn_apply_kernel(
    const float* __restrict__ Y, const float* __restrict__ scale,
    const float* __restrict__ shift, float* __restrict__ out, long total) {
  long i = (long)blockIdx.x * 256 + threadIdx.x;
  if (i >= total) return;
  int col = (int)(i & (EMB - 1));
  out[i] = Y[i] * scale[col] + shift[col];
}

// ---------------------------------------------------------------------------
// Kernel 7: H = gelu_exact(X @ W1(128x512) + b1). grid.y picks 128-col slab.
// ---------------------------------------------------------------------------
__global__ __launch_bounds__(256) void ff1_kernel(
    const float* __restrict__ X, const float* __restrict__ W1,
    const float* __restrict__ b1, float* __restrict__ Hh) {
  __shared__ __align__(16) float xs[16 * LDS_STRIDE];
  const int m0 = blockIdx.x * 16;
  const int t  = threadIdx.x;
  for (int i = t; i < 16 * EMB; i += 256) {
    int r = i >> 7, c = i & 127;
    xs[r * LDS_STRIDE + c] = X[(long)(m0 + r) * EMB + c];
  }
  __syncthreads();
  const int wave = t >> 5, lane = t & 31;
  const int n = blockIdx.y * 128 + wave * 16 + (lane & 15);
  const int m = lane & 15;
  const int ksel = (lane >> 4) * 2;
  v8f c = {};
  for (int kb = 0; kb < EMB; kb += 4) {
    v2f a = *(const v2f*)(xs + m * LDS_STRIDE + kb + ksel);
    const int kr = kb + ksel;
    v2f b;
    b[0] = W1[kr * HID + n]; b[1] = W1[(kr + 1) * HID + n];
    c = wmma_f32(a, b, c);
  }
  const float bb = b1[n];
  const int rbase = m0 + (lane >> 4) * 8;
  for (int j = 0; j < 8; ++j) {
    float y = c[j] + bb;
    y = 0.5f * y * (1.0f + erff(y * 0.70710678118654752440f));  // exact gelu
    Hh[(long)(rbase + j) * HID + n] = y;
  }
}

// ---------------------------------------------------------------------------
// Kernel 8: Y = H @ W2(512x128) + b2 + resid, column stats (K chunked by 128)
// ---------------------------------------------------------------------------
__global__ __launch_bounds__(256) void ff2_kernel(
    const float* __restrict__ Hh, const float* __restrict__ W2,
    const float* __restrict__ b2, const float* __restrict__ resid,
    float* __restrict__ Y, float* __restrict__ csum, float* __restrict__ csq) {
  __shared__ __align__(16) float xs[16 * LDS_STRIDE];
  const int m0 = blockIdx.x * 16;
  const int t  = threadIdx.x;
  const int wave = t >> 5, lane = t & 31;
  const int n = wave * 16 + (lane & 15);
  const int m = lane & 15;
  const int ksel = (lane >> 4) * 2;
  v8f c = {};
  for (int kc = 0; kc < HID; kc += EMB) {
    __syncthreads();
    for (int i = t; i < 16 * EMB; i += 256) {
      int r = i >> 7, cc = i & 127;
      xs[r * LDS_STRIDE + cc] = Hh[(long)(m0 + r) * HID + kc + cc];
    }
    __syncthreads();
    for (int kb = 0; kb < EMB; kb += 4) {
      v2f a = *(const v2f*)(xs + m * LDS_STRIDE + kb + ksel);
      const int kr = kc + kb + ksel;
      v2f b;
      b[0] = W2[kr * EMB + n]; b[1] = W2[(kr + 1) * EMB + n];
      c = wmma_f32(a, b, c);
    }
  }
  const float bb = b2[n];
  const int rbase = m0 + (lane >> 4) * 8;
  float lsum = 0.0f, lsq = 0.0f;
  for (int j = 0; j < 8; ++j) {
    long idx = (long)(rbase + j) * EMB + n;
    float y = c[j] + bb + resid[idx];
    Y[idx] = y;
    lsum += y; lsq += y * y;
  }
  ATOMIC_ADD_F32(csum + n, lsum);
  ATOMIC_ADD_F32(csq + n, lsq);
}

// ---------------------------------------------------------------------------
extern "C" void kernel_launch(void* const* d_in, const int* in_sizes, int n_in,
                              void* d_out, int out_size, void* d_ws, size_t ws_size,
                              hipStream_t stream) {
  const float* x   = (const float*)d_in[0];
  const float* wq  = (const float*)d_in[1];  const float* bq = (const float*)d_in[2];
  const float* wk  = (const float*)d_in[3];  const float* bk = (const float*)d_in[4];
  const float* wv  = (const float*)d_in[5];  const float* bv = (const float*)d_in[6];
  const float* wo  = (const float*)d_in[7];  const float* bo = (const float*)d_in[8];
  const float* w1  = (const float*)d_in[9];  const float* b1 = (const float*)d_in[10];
  const float* w2  = (const float*)d_in[11]; const float* b2 = (const float*)d_in[12];
  const float* g1  = (const float*)d_in[13]; const float* be1 = (const float*)d_in[14];
  const float* g2  = (const float*)d_in[15]; const float* be2 = (const float*)d_in[16];
  const int*   ei  = (const int*)d_in[17];
  float* out = (float*)d_out;

  // ---- workspace layout (floats) ----
  const long NF = (long)NN * EMB;              // 6.4M floats
  float* ws = (float*)d_ws;
  float* q    = ws;                            // [N,128]
  float* k    = q    + NF;                     // [N,128]
  float* v    = k    + NF;                     // [N,128]
  float* ksum = v    + NF;                     // [N,128]
  float* cnt  = ksum + NF;                     // [N] (padded)
  float* attn = cnt  + 50048;                  // [N,128]
  float* y1   = attn + NF;                     // [N,128]
  float* x1   = y1   + NF;                     // [N,128]
  float* y2   = x1   + NF;                     // [N,128]
  float* st   = y2   + NF;                     // stats: 8 x 128
  float* csum1 = st;        float* csq1 = st + 128;
  float* csum2 = st + 256;  float* csq2 = st + 384;
  float* scale1 = st + 512; float* shift1 = st + 640;
  float* scale2 = st + 768; float* shift2 = st + 896;
  float* hbuf = ws;  // [N,512] aliases dead q/k/v/ksum after attn_kernel

  const int MT = NN / 16;                      // 3125 M-tiles
  const float invn = 1.0f / (float)NN;

  // zero scatter targets + stats (re-zeroed every call: graph-replay safe)
  zero_kernel<<<4096, 256, 0, stream>>>(ksum, NF + 50048);
  zero_kernel<<<2, 256, 0, stream>>>(st, 512);

  qkv_kernel<<<MT, 256, 0, stream>>>(x, wq, bq, wk, bk, wv, bv, q, k, v);
  edge_kernel<<<(EE * 32) / 256, 256, 0, stream>>>(ei, k, ksum, cnt);
  attn_kernel<<<NN, 128, 0, stream>>>(q, ksum, v, cnt, attn);
  proj_res_kernel<<<MT, 256, 0, stream>>>(attn, wo, bo, x, y1, csum1, csq1);
  bn_finalize_kernel<<<1, 128, 0, stream>>>(csum1, csq1, g1, be1, scale1, shift1, invn);
  bn_apply_kernel<<<(int)(NF / 256), 256, 0, stream>>>(y1, scale1, shift1, x1, NF);
  ff1_kernel<<<dim3(MT, 4), 256, 0, stream>>>(x1, w1, b1, hbuf);
  ff2_kernel<<<MT, 256, 0, stream>>>(hbuf, w2, b2, x1, y2, csum2, csq2);
  bn_finalize_kernel<<<1, 128, 0, stream>>>(csum2, csq2, g2, be2, scale2, shift2, invn);
  bn_apply_kernel<<<(int)(NF / 256), 256, 0, stream>>>(y2, scale2, shift2, out, NF);
}